// HoT_GNN_87385404604873
// MI455X (gfx1250) — compile-verified
//
#include <hip/hip_runtime.h>
#include <math.h>

typedef float v2f __attribute__((ext_vector_type(2)));
typedef float v8f __attribute__((ext_vector_type(8)));

#define N_NODES 6144
#define N_EDGES 12288
#define WAVES 8

// D = A(16x4 f32) x B(4x16 f32) + C(16x16 f32)
__device__ __forceinline__ v8f wmma4(v2f a, v2f b, v8f c) {
  return __builtin_amdgcn_wmma_f32_16x16x4_f32(false, a, false, b, (short)0, c,
                                               false, false);
}

// ---------------------------------------------------------------------------
// GNN layer: Hout = relu((A @ Xin) @ W^T + bias), Xin:[K,32], W:[32,32]
// One block = one 16-row tile. 8 waves split K; LDS reduce; WMMA epilogue.
// ---------------------------------------------------------------------------
__global__ __launch_bounds__(256) void gnn_kernel(
    const float* __restrict__ A, const float* __restrict__ Xin,
    const float* __restrict__ W, const float* __restrict__ bias,
    float* __restrict__ Hout, int K) {
  __shared__ float lds_part[WAVES * 512];
  __shared__ float lds_T[512];
  __shared__ float lds_W[1024];

  const int tid = threadIdx.x;
  const int wave = tid >> 5;
  const int lane = tid & 31;
  const int l15 = lane & 15;
  const int hi = lane >> 4;  // 0: K parts {0,1}; 1: K parts {2,3}
  const int mbase = blockIdx.x * 16;

  for (int i = tid; i < 1024; i += 256) lds_W[i] = W[i];

  const int kchunk = K / WAVES;
  const int k0 = wave * kchunk;

  const float* aptr = A + (size_t)(mbase + l15) * K + k0 + hi * 2;
  const float* bptr = Xin + (size_t)(k0 + hi * 2) * 32;

  v8f c0 = {};
  v8f c1 = {};
  for (int k = 0; k < kchunk; k += 4) {
    v2f a = *(const v2f*)aptr;
    aptr += 4;
    v2f b0, b1;
    b0.x = bptr[l15];
    b0.y = bptr[32 + l15];
    b1.x = bptr[16 + l15];
    b1.y = bptr[48 + l15];
    bptr += 128;
    c0 = wmma4(a, b0, c0);
    c1 = wmma4(a, b1, c1);
  }

  // store per-wave partial 16x32 tile
  float* p = lds_part + wave * 512;
#pragma unroll
  for (int r = 0; r < 8; r++) {
    const int row = r + hi * 8;
    p[row * 32 + l15] = c0[r];
    p[row * 32 + l15 + 16] = c1[r];
  }
  __syncthreads();
  for (int i = tid; i < 512; i += 256) {
    float s = 0.f;
#pragma unroll
    for (int w = 0; w < WAVES; w++) s += lds_part[w * 512 + i];
    lds_T[i] = s;
  }
  __syncthreads();

  if (wave == 0) {
    // epilogue: D = T(16x32) @ W^T(32x32), then bias+relu.  B[j][c] = W[c][j]
    v8f d0 = {};
    v8f d1 = {};
#pragma unroll
    for (int j = 0; j < 32; j += 4) {
      v2f a = *(const v2f*)(&lds_T[l15 * 32 + j + hi * 2]);
      v2f b0 = *(const v2f*)(&lds_W[l15 * 32 + j + hi * 2]);
      v2f b1 = *(const v2f*)(&lds_W[(l15 + 16) * 32 + j + hi * 2]);
      d0 = wmma4(a, b0, d0);
      d1 = wmma4(a, b1, d1);
    }
    const float bias0 = bias[l15];
    const float bias1 = bias[l15 + 16];
#pragma unroll
    for (int r = 0; r < 8; r++) {
      const int row = r + hi * 8;
      float x0 = d0[r] + bias0;
      float x1 = d1[r] + bias1;
      x0 = x0 > 0.f ? x0 : 0.f;
      x1 = x1 > 0.f ? x1 : 0.f;
      Hout[(size_t)(mbase + row) * 32 + l15] = x0;
      Hout[(size_t)(mbase + row) * 32 + l15 + 16] = x1;
    }
  }
}

// ---------------------------------------------------------------------------
// Zt0[e][j] = sum_i Xe[e][i]*W0[j][i] + b0[j]
// ---------------------------------------------------------------------------
__global__ __launch_bounds__(256) void theta0_kernel(
    const float* __restrict__ Xe, const float* __restrict__ W,
    const float* __restrict__ b, float* __restrict__ Zt) {
  const int e = blockIdx.x * 256 + threadIdx.x;
  if (e >= N_EDGES) return;
  float x[16];
#pragma unroll
  for (int i = 0; i < 16; i++) x[i] = Xe[(size_t)e * 16 + i];
#pragma unroll
  for (int j = 0; j < 16; j++) {
    float acc = b[j];
#pragma unroll
    for (int i = 0; i < 16; i++) acc += x[i] * W[j * 16 + i];
    Zt[(size_t)e * 16 + j] = acc;
  }
}

// ---------------------------------------------------------------------------
// Zc = L1 @ Zt (Zt:[K,16]) + per-row sums of L1 (fused, free from A-frags)
// ---------------------------------------------------------------------------
__global__ __launch_bounds__(256) void hosc_conv_kernel(
    const float* __restrict__ L1, const float* __restrict__ Zt,
    float* __restrict__ Zc, float* __restrict__ rowsum, int K) {
  __shared__ float lds_part[WAVES * 256];
  __shared__ float lds_rs[WAVES * 32];

  const int tid = threadIdx.x;
  const int wave = tid >> 5;
  const int lane = tid & 31;
  const int l15 = lane & 15;
  const int hi = lane >> 4;
  const int mbase = blockIdx.x * 16;

  const int kchunk = K / WAVES;
  const int k0 = wave * kchunk;

  const float* aptr = L1 + (size_t)(mbase + l15) * K + k0 + hi * 2;
  const float* bptr = Zt + (size_t)(k0 + hi * 2) * 16;

  v8f c = {};
  float rs = 0.f;
  for (int k = 0; k < kchunk; k += 4) {
    v2f a = *(const v2f*)aptr;
    aptr += 4;
    rs += a.x + a.y;
    v2f b;
    b.x = bptr[l15];
    b.y = bptr[16 + l15];
    bptr += 64;
    c = wmma4(a, b, c);
  }

  float* p = lds_part + wave * 256;
#pragma unroll
  for (int r = 0; r < 8; r++) p[(r + hi * 8) * 16 + l15] = c[r];
  lds_rs[wave * 32 + lane] = rs;
  __syncthreads();

  {  // reduce 16x16 tile (256 elements == 256 threads)
    float s = 0.f;
#pragma unroll
    for (int w = 0; w < WAVES; w++) s += lds_part[w * 256 + tid];
    const int row = tid >> 4, col = tid & 15;
    Zc[(size_t)(mbase + row) * 16 + col] = s;
  }
  if (tid < 16) {
    float s = 0.f;
#pragma unroll
    for (int w = 0; w < WAVES; w++)
      s += lds_rs[w * 32 + tid] + lds_rs[w * 32 + tid + 16];
    rowsum[mbase + tid] = s;
  }
}

// ---------------------------------------------------------------------------
// BN stats over columns of Zc[rows,16]: stats = {mean[16], rsqrt(var+eps)[16]}
// ---------------------------------------------------------------------------
__global__ __launch_bounds__(256) void bnstats_kernel(
    const float* __restrict__ Zc, float* __restrict__ stats, int rows) {
  __shared__ float ssum[256], ssq[256];
  const int tid = threadIdx.x;
  const int ch = tid & 15, grp = tid >> 4;
  float s = 0.f, q = 0.f;
  for (int r = grp; r < rows; r += 16) {
    const float v = Zc[(size_t)r * 16 + ch];
    s += v;
    q += v * v;
  }
  ssum[tid] = s;
  ssq[tid] = q;
  __syncthreads();
  if (tid < 16) {
    float S = 0.f, Q = 0.f;
#pragma unroll
    for (int g = 0; g < 16; g++) {
      S += ssum[g * 16 + tid];
      Q += ssq[g * 16 + tid];
    }
    const float m = S / (float)rows;
    const float v = Q / (float)rows - m * m;
    stats[tid] = m;
    stats[16 + tid] = rsqrtf(v + 1e-5f);
  }
}

// ---------------------------------------------------------------------------
// Z0[e] = max_j relu(g[j]*(Zc[e,j]-m[j])*rs[j] + beta[j])
// ---------------------------------------------------------------------------
__global__ __launch_bounds__(256) void zmax0_kernel(
    const float* __restrict__ Zc, const float* __restrict__ stats,
    const float* __restrict__ g, const float* __restrict__ beta,
    float* __restrict__ Z0, int rows) {
  const int e = blockIdx.x * 256 + threadIdx.x;
  if (e >= rows) return;
  float mx = 0.f;  // relu outputs are >= 0
#pragma unroll
  for (int j = 0; j < 16; j++) {
    float v = g[j] * (Zc[(size_t)e * 16 + j] - stats[j]) * stats[16 + j] + beta[j];
    v = v > 0.f ? v : 0.f;
    mx = v > mx ? v : mx;
  }
  Z0[e] = mx;
}

// ---------------------------------------------------------------------------
// y[row] = dot(M[row,:], x)  (one block per row)
// ---------------------------------------------------------------------------
__global__ __launch_bounds__(256) void matvec_kernel(
    const float* __restrict__ M, const float* __restrict__ x,
    float* __restrict__ y, int K) {
  __shared__ float red[256];
  const int row = blockIdx.x;
  const float* r = M + (size_t)row * K;
  float acc = 0.f;
  for (int i = threadIdx.x; i < K; i += 256) acc += r[i] * x[i];
  red[threadIdx.x] = acc;
  __syncthreads();
  for (int s = 128; s > 0; s >>= 1) {
    if (threadIdx.x < s) red[threadIdx.x] += red[threadIdx.x + s];
    __syncthreads();
  }
  if (threadIdx.x == 0) y[row] = red[0];
}

__global__ __launch_bounds__(256) void matvec2_kernel(
    const float* __restrict__ M, const float* __restrict__ x0,
    const float* __restrict__ x1, float* __restrict__ y0,
    float* __restrict__ y1, int K) {
  __shared__ float r0[256], r1[256];
  const int row = blockIdx.x;
  const float* r = M + (size_t)row * K;
  float a0 = 0.f, a1 = 0.f;
  for (int i = threadIdx.x; i < K; i += 256) {
    const float m = r[i];
    a0 += m * x0[i];
    a1 += m * x1[i];
  }
  r0[threadIdx.x] = a0;
  r1[threadIdx.x] = a1;
  __syncthreads();
  for (int s = 128; s > 0; s >>= 1) {
    if (threadIdx.x < s) {
      r0[threadIdx.x] += r0[threadIdx.x + s];
      r1[threadIdx.x] += r1[threadIdx.x + s];
    }
    __syncthreads();
  }
  if (threadIdx.x == 0) {
    y0[row] = r0[0];
    y1[row] = r1[0];
  }
}

// ---------------------------------------------------------------------------
// BN stats for layer-1 via (u,s) moments: Zc1[:,j] = u*W1[j] + s*b1[j]
// st = {m1[16], rsqrt(v1+eps)[16]}
// ---------------------------------------------------------------------------
__global__ __launch_bounds__(256) void stats2_kernel(
    const float* __restrict__ u, const float* __restrict__ s,
    const float* __restrict__ W1, const float* __restrict__ b1,
    float* __restrict__ st, int rows) {
  __shared__ float r0[256], r1[256], r2[256], r3[256], r4[256];
  const int tid = threadIdx.x;
  float su = 0.f, squ = 0.f, ss = 0.f, sqs = 0.f, sus = 0.f;
  for (int i = tid; i < rows; i += 256) {
    const float uu = u[i], sv = s[i];
    su += uu;
    squ += uu * uu;
    ss += sv;
    sqs += sv * sv;
    sus += uu * sv;
  }
  r0[tid] = su; r1[tid] = squ; r2[tid] = ss; r3[tid] = sqs; r4[tid] = sus;
  __syncthreads();
  for (int st2 = 128; st2 > 0; st2 >>= 1) {
    if (tid < st2) {
      r0[tid] += r0[tid + st2];
      r1[tid] += r1[tid + st2];
      r2[tid] += r2[tid + st2];
      r3[tid] += r3[tid + st2];
      r4[tid] += r4[tid + st2];
    }
    __syncthreads();
  }
  if (tid < 16) {
    const float inv = 1.f / (float)rows;
    const float mu_u = r0[0] * inv, mu_s = r2[0] * inv;
    const float var_u = r1[0] * inv - mu_u * mu_u;
    const float var_s = r3[0] * inv - mu_s * mu_s;
    const float cov = r4[0] * inv - mu_u * mu_s;
    const float w = W1[tid], bb = b1[tid];
    st[tid] = mu_u * w + mu_s * bb;
    const float v1 = w * w * var_u + bb * bb * var_s + 2.f * w * bb * cov;
    st[16 + tid] = rsqrtf(v1 + 1e-5f);
  }
}

__global__ __launch_bounds__(256) void zmax1_kernel(
    const float* __restrict__ u, const float* __restrict__ s,
    const float* __restrict__ W1, const float* __restrict__ b1,
    const float* __restrict__ st, const float* __restrict__ g,
    const float* __restrict__ beta, float* __restrict__ Z1, int rows) {
  const int e = blockIdx.x * 256 + threadIdx.x;
  if (e >= rows) return;
  const float uu = u[e], sv = s[e];
  float mx = 0.f;
#pragma unroll
  for (int j = 0; j < 16; j++) {
    const float zc = uu * W1[j] + sv * b1[j];
    float v = g[j] * (zc - st[j]) * st[16 + j] + beta[j];
    v = v > 0.f ? v : 0.f;
    mx = v > mx ? v : mx;
  }
  Z1[e] = mx;
}

// ---------------------------------------------------------------------------
// Head: out[n] = sigmoid( sum_h relu(dot(Hcat[n],fc1W[h]) + fc1b[h]) * fc2W[h]
//                         + fc2b )  with Hcat = [H1 | He0 | He1], K=34 pad 36.
// Wave handles 16 nodes; A-fragments held in registers across hidden loop.
// ---------------------------------------------------------------------------
__global__ __launch_bounds__(256) void head_kernel(
    const float* __restrict__ H1, const float* __restrict__ He0,
    const float* __restrict__ He1, const float* __restrict__ fc1W,
    const float* __restrict__ fc1b, const float* __restrict__ fc2W,
    const float* __restrict__ fc2b, float* __restrict__ out) {
  const int tid = threadIdx.x;
  const int wave = tid >> 5;
  const int lane = tid & 31;
  const int l15 = lane & 15;
  const int hi = lane >> 4;
  const float mlo = (hi == 0) ? 1.f : 0.f;  // mask: hi half covers padded K
  const int node_base = blockIdx.x * 128 + wave * 16;

  v2f afrag[9];
  const float* hrow = H1 + (size_t)(node_base + l15) * 32;
#pragma unroll
  for (int j = 0; j < 8; j++) afrag[j] = *(const v2f*)(hrow + j * 4 + hi * 2);
  {
    v2f a8;
    a8.x = He0[node_base + l15] * mlo;
    a8.y = He1[node_base + l15] * mlo;
    afrag[8] = a8;
  }

  float acc[8];
#pragma unroll
  for (int r = 0; r < 8; r++) acc[r] = 0.f;

  for (int hc = 0; hc < 2 * N_NODES; hc += 16) {
    const float* wrow = fc1W + (size_t)(hc + l15) * 34;
    v8f c = {};
#pragma unroll
    for (int j = 0; j < 8; j++) {
      v2f b = *(const v2f*)(wrow + j * 4 + hi * 2);
      c = wmma4(afrag[j], b, c);
    }
    v2f b8;
    b8.x = wrow[32] * mlo;
    b8.y = wrow[33] * mlo;
    c = wmma4(afrag[8], b8, c);

    const float fb = fc1b[hc + l15];
    const float w2 = fc2W[hc + l15];
#pragma unroll
    for (int r = 0; r < 8; r++) {
      float v = c[r] + fb;
      v = v > 0.f ? v : 0.f;
      acc[r] += v * w2;
    }
  }

  // reduce 16 columns per row; lanes 0-15 hold rows 0-7, lanes 16-31 rows 8-15
#pragma unroll
  for (int r = 0; r < 8; r++) {
    float v = acc[r];
    for (int off = 1; off < 16; off <<= 1) v += __shfl_xor(v, off, 16);
    if (l15 == 0) {
      const int node = node_base + r + hi * 8;
      const float x = v + fc2b[0];
      out[node] = 1.f / (1.f + expf(-x));
    }
  }
}

// ---------------------------------------------------------------------------
extern "C" void kernel_launch(void* const* d_in, const int* in_sizes, int n_in,
                              void* d_out, int out_size, void* d_ws,
                              size_t ws_size, hipStream_t stream) {
  (void)in_sizes; (void)n_in; (void)out_size; (void)ws_size;

  const float* X_n    = (const float*)d_in[0];
  const float* X_e    = (const float*)d_in[1];
  const float* A      = (const float*)d_in[2];
  const float* L1     = (const float*)d_in[3];
  const float* B1     = (const float*)d_in[4];
  const float* gnn_W0 = (const float*)d_in[5];
  const float* gnn_b0 = (const float*)d_in[6];
  const float* gnn_W1 = (const float*)d_in[7];
  const float* gnn_b1 = (const float*)d_in[8];
  const float* th_W0  = (const float*)d_in[9];
  const float* th_b0  = (const float*)d_in[10];
  const float* bn_g0  = (const float*)d_in[11];
  const float* bn_b0  = (const float*)d_in[12];
  const float* th_W1  = (const float*)d_in[13];
  const float* th_b1  = (const float*)d_in[14];
  const float* bn_g1  = (const float*)d_in[15];
  const float* bn_b1  = (const float*)d_in[16];
  const float* fc1_W  = (const float*)d_in[17];
  const float* fc1_b  = (const float*)d_in[18];
  const float* fc2_W  = (const float*)d_in[19];
  const float* fc2_b  = (const float*)d_in[20];
  float* out = (float*)d_out;

  float* ws = (float*)d_ws;
  float* H0  = ws;                 // N*32
  float* H1  = H0 + N_NODES * 32;  // N*32
  float* Zt0 = H1 + N_NODES * 32;  // E*16
  float* Zc0 = Zt0 + N_EDGES * 16; // E*16
  float* srs = Zc0 + N_EDGES * 16; // E  (rowsums of L1)
  float* u   = srs + N_EDGES;      // E
  float* Z0  = u + N_EDGES;        // E
  float* Z1  = Z0 + N_EDGES;       // E
  float* st0 = Z1 + N_EDGES;       // 32
  float* st1 = st0 + 32;           // 32
  float* He0 = st1 + 32;           // N
  float* He1 = He0 + N_NODES;      // N

  // GNN (two layers, same WMMA kernel)
  gnn_kernel<<<N_NODES / 16, 256, 0, stream>>>(A, X_n, gnn_W0, gnn_b0, H0, N_NODES);
  gnn_kernel<<<N_NODES / 16, 256, 0, stream>>>(A, H0, gnn_W1, gnn_b1, H1, N_NODES);

  // HoSC layer 0
  theta0_kernel<<<N_EDGES / 256, 256, 0, stream>>>(X_e, th_W0, th_b0, Zt0);
  hosc_conv_kernel<<<N_EDGES / 16, 256, 0, stream>>>(L1, Zt0, Zc0, srs, N_EDGES);
  bnstats_kernel<<<1, 256, 0, stream>>>(Zc0, st0, N_EDGES);
  zmax0_kernel<<<N_EDGES / 256, 256, 0, stream>>>(Zc0, st0, bn_g0, bn_b0, Z0, N_EDGES);

  // HoSC layer 1 collapsed to matvec (rank-1 structure)
  matvec_kernel<<<N_EDGES, 256, 0, stream>>>(L1, Z0, u, N_EDGES);
  stats2_kernel<<<1, 256, 0, stream>>>(u, srs, th_W1, th_b1, st1, N_EDGES);
  zmax1_kernel<<<N_EDGES / 256, 256, 0, stream>>>(u, srs, th_W1, th_b1, st1,
                                                  bn_g1, bn_b1, Z1, N_EDGES);

  // H_e = B1 @ [Z0, Z1]
  matvec2_kernel<<<N_NODES, 256, 0, stream>>>(B1, Z0, Z1, He0, He1, N_EDGES);

  // fused head (WMMA)
  head_kernel<<<N_NODES / 128, 256, 0, stream>>>(H1, He0, He1, fc1_W, fc1_b,
                                                 fc2_W, fc2_b, out);
}